// STAttn_9594956939719
// MI455X (gfx1250) — compile-verified
//
#include <hip/hip_runtime.h>
#include <hip/hip_bf16.h>

// CDNA5 / gfx1250 fused STAttn:
//   per-bt workgroup: TDM-load x tile -> WMMA f32 16x16x4 GEMM (h = x @ ue_w^T)
//   -> leaky_relu -> score matvec -> softmax(N) -> weighted pooling -> fc matvec.
// B=32, T=32, N=64, D=512, H=64, OUT=256.

typedef __attribute__((ext_vector_type(2))) float        v2f;
typedef __attribute__((ext_vector_type(8))) float        v8f;
typedef __attribute__((ext_vector_type(4))) unsigned int v4u;
typedef __attribute__((ext_vector_type(8))) int          v8i;
typedef __attribute__((ext_vector_type(4))) int          v4i;

// x tile in LDS: 64 rows x 512 f32, TDM pads 2 DWORDs after every 256 DWORDs
// => row stride 516 DWORDs, intra-row +2 gap at d>=256. This makes the
// 16-lane WMMA A-fragment loads hit 32 distinct LDS banks (4*r + k pattern).
#define XSTRIDE 516
__device__ __forceinline__ int xoff(int n, int d) {
  return n * XSTRIDE + d + ((d & 256) ? 2 : 0);
}

__global__ __launch_bounds__(512) void stattn_fused(
    const float* __restrict__ inputs,   // [1024, 64, 512]
    const float* __restrict__ ue_w,     // [64, 512]
    const float* __restrict__ ue_b,     // [64]
    const float* __restrict__ be,       // [64]
    const float* __restrict__ w_w,      // [1, 64]
    const float* __restrict__ w_b,      // [1]
    const float* __restrict__ fc1_w,    // [256, 512]
    const float* __restrict__ fc1_b,    // [256]
    float* __restrict__ out)            // [32, 32, 256] = [T, B, OUT]
{
  __shared__ float xs[64 * XSTRIDE];    // 132096 B : x tile (padded)
  __shared__ float hs[64 * 68];         //  17408 B : h = leaky(x@ue_w^T + b)
  __shared__ float ev[64];              // scores
  __shared__ float aw[64];              // softmax weights
  __shared__ float attr[512];           // pooled vector

  const int bt   = blockIdx.x;          // bt = b*T + t
  const int tid  = threadIdx.x;
  const int wv   = tid >> 5;            // wave id 0..15
  const int lane = tid & 31;
  const int l    = lane & 15;
  const int half = lane >> 4;

  // ---------------- Stage 1: TDM async tensor load of x[bt] into LDS -------
  if (tid < 32) {                       // wave 0 only (TDM ignores EXEC)
    unsigned long long ga =
        (unsigned long long)(const void*)(inputs + (size_t)bt * 64 * 512);
    unsigned lds_base = (unsigned)(unsigned long long)(void*)xs;

    // D# group 0: count=1 | lds_addr | global_addr | type=2
    v4u g0 = { 1u,
               lds_base,
               (unsigned)(ga & 0xFFFFFFFFull),
               (unsigned)((ga >> 32) & 0x01FFFFFFull) | 0x80000000u };
    // D# group 1:
    //  w0: data_size=2 (4B) | pad_enable | pad_interval=7 (256 dw) | pad_amount=1 (2 dw)
    //  w1[31:16]=tensor_dim0.lo(512)  w2[31:16]=tensor_dim1.lo(64)
    //  w3[31:16]=tile_dim0(512)       w4[15:0]=tile_dim1(64)
    //  w5=tensor_dim0_stride(512)
    v8i g1 = { (int)((2u << 16) | (1u << 20) | (7u << 22) | (1u << 25)),
               (int)(512u << 16),
               (int)(64u  << 16),
               (int)(512u << 16),
               64,
               512,
               0, 0 };
    v4i zz4 = {0, 0, 0, 0};
    v8i zz8 = {0, 0, 0, 0, 0, 0, 0, 0};
    __builtin_amdgcn_tensor_load_to_lds(g0, g1, zz4, zz4, zz8, 0);
    __builtin_amdgcn_s_wait_tensorcnt(0);
  }
  __syncthreads();

  // ---------------- Stage 2: h = leaky_relu(x @ ue_w^T + ue_b + be) --------
  // 16 waves, wave (mi,ni) computes tile rows mi*16.., cols ni*16..
  {
    const int mi   = wv >> 2;
    const int ni   = wv & 3;
    const int arow = mi * 16 + l;            // x row (M)
    const int col  = ni * 16 + l;            // h col (N)
    const float* wptr = ue_w + (size_t)col * 512 + 2 * half;

    v8f acc0 = {0.f, 0.f, 0.f, 0.f, 0.f, 0.f, 0.f, 0.f};
    v8f acc1 = {0.f, 0.f, 0.f, 0.f, 0.f, 0.f, 0.f, 0.f};
#pragma unroll 4
    for (int k = 0; k < 512; k += 8) {
      // A fragment: lanes 0-15 hold K=k..k+1, lanes 16-31 K=k+2..k+3
      v2f a0 = *(const v2f*)&xs[xoff(arow, k + 2 * half)];
      v2f b0 = *(const v2f*)&wptr[k];
      acc0 = __builtin_amdgcn_wmma_f32_16x16x4_f32(
          false, a0, false, b0, (short)0, acc0, false, false);
      v2f a1 = *(const v2f*)&xs[xoff(arow, k + 4 + 2 * half)];
      v2f b1 = *(const v2f*)&wptr[k + 4];
      acc1 = __builtin_amdgcn_wmma_f32_16x16x4_f32(
          false, a1, false, b1, (short)0, acc1, false, false);
    }
    const float bias = ue_b[col] + be[col];
#pragma unroll
    for (int r = 0; r < 8; ++r) {
      float v = acc0[r] + acc1[r] + bias;
      v = v > 0.0f ? v : 0.2f * v;           // leaky_relu(0.2)
      hs[(mi * 16 + r + 8 * half) * 68 + col] = v;  // row = r + 8*(lane>=16)
    }
  }
  __syncthreads();

  // ---------------- Stage 3: scores e[n] = h[n,:]·w_w + w_b, softmax over N
  if (tid < 64) {
    float s = w_b[0];
#pragma unroll 8
    for (int hh = 0; hh < 64; ++hh) s += hs[tid * 68 + hh] * w_w[hh];
    ev[tid] = s;
  }
  __syncthreads();
  if (tid < 64) {
    float mx = ev[0];
    for (int n = 1; n < 64; ++n) mx = fmaxf(mx, ev[n]);
    float sum = 0.0f;
    for (int n = 0; n < 64; ++n) sum += expf(ev[n] - mx);
    aw[tid] = expf(ev[tid] - mx) / sum;
  }
  __syncthreads();

  // ---------------- Stage 4: attr[d] = sum_n aw[n] * x[n,d] ----------------
  {
    float acc = 0.0f;
#pragma unroll 8
    for (int n = 0; n < 64; ++n) acc += aw[n] * xs[xoff(n, tid)];
    attr[tid] = acc;
  }
  __syncthreads();

  // ---------------- Stage 5: fc[o] = attr · fc1_w[o,:] + fc1_b[o] ----------
  // wave wv handles outputs o = wv*16 .. wv*16+15; lane-parallel dot + shuffle
  {
    const int t = bt & 31;                   // bt = b*32 + t
    const int b = bt >> 5;
    float* orow = out + ((size_t)t * 32 + b) * 256;
    for (int j = 0; j < 16; ++j) {
      const int o = wv * 16 + j;
      const float* fw = fc1_w + (size_t)o * 512;
      float p = 0.0f;
#pragma unroll
      for (int i = 0; i < 16; ++i) {
        const int d = lane + 32 * i;
        p += attr[d] * fw[d];
      }
#pragma unroll
      for (int s = 16; s > 0; s >>= 1) p += __shfl_down(p, s, 32);
      if (lane == 0) orow[o] = p + fc1_b[o];
    }
  }
}

extern "C" void kernel_launch(void* const* d_in, const int* in_sizes, int n_in,
                              void* d_out, int out_size, void* d_ws, size_t ws_size,
                              hipStream_t stream) {
  (void)in_sizes; (void)n_in; (void)d_ws; (void)ws_size; (void)out_size;
  const float* inputs = (const float*)d_in[0];
  const float* ue_w   = (const float*)d_in[1];
  const float* ue_b   = (const float*)d_in[2];
  const float* be     = (const float*)d_in[3];
  const float* w_w    = (const float*)d_in[4];
  const float* w_b    = (const float*)d_in[5];
  const float* fc1_w  = (const float*)d_in[6];
  const float* fc1_b  = (const float*)d_in[7];
  float* out = (float*)d_out;

  stattn_fused<<<dim3(32 * 32), dim3(512), 0, stream>>>(
      inputs, ue_w, ue_b, be, w_w, w_b, fc1_w, fc1_b, out);
}